// D4RTLoss_89300960019127
// MI455X (gfx1250) — compile-verified
//
#include <hip/hip_runtime.h>
#include <hip/hip_bf16.h>

// Problem constants (from reference): B=16, N=131072, 16 groups.
#define B_    16
#define N_    131072
#define G_    16
#define P_    (B_ * N_)          // 2,097,152 points
#define BG_   256                // B*G group slots
#define BINS1 1024               // key bits [31:22]
#define BINS2 2048               // key bits [21:11]
#define BINS3 2048               // key bits [10:0]

// Workspace layout (uint32 word offsets)
#define OFF_H1   0u
#define OFF_H2   (OFF_H1 + BG_ * BINS1)      //   262144
#define OFF_H3   (OFF_H2 + BG_ * BINS2)      //   786432
#define OFF_CNT  (OFF_H3 + BG_ * BINS3)      //  1310720
#define OFF_SEL1 (OFF_CNT + 256u)
#define OFF_REM1 (OFF_SEL1 + 256u)
#define OFF_PFX  (OFF_REM1 + 256u)
#define OFF_REM2 (OFF_PFX + 256u)
#define OFF_INV  (OFF_REM2 + 256u)
#define OFF_BS   (OFF_INV + 256u)            // float[2048] per-block loss sums
#define ZERO_U4  ((OFF_CNT) / 4u)            // 327680 uint4 -> exactly 1280 blocks * 256

typedef float v2f __attribute__((ext_vector_type(2)));
typedef float v8f __attribute__((ext_vector_type(8)));

// Monotonic uint key: order-preserving map of float -> uint32 (ascending).
__device__ __forceinline__ unsigned int fkey(float f) {
  unsigned int u = __float_as_uint(f);
  return (u & 0x80000000u) ? ~u : (u | 0x80000000u);
}
__device__ __forceinline__ float fval(unsigned int k) {
  unsigned int u = (k & 0x80000000u) ? (k ^ 0x80000000u) : ~k;
  return __uint_as_float(u);
}
__device__ __forceinline__ float logt(float x) {
  return __builtin_copysignf(log1pf(fabsf(x)), x);
}

__global__ __launch_bounds__(256) void zero_ws_kernel(uint4* __restrict__ ws) {
  unsigned int i = blockIdx.x * 256u + threadIdx.x;   // grid sized exactly
  ws[i] = make_uint4(0u, 0u, 0u, 0u);
}

// Pass 1: coarse histogram (top 10 key bits) per (batch, group) via 64KB LDS hist.
__global__ __launch_bounds__(256) void hist1_kernel(const float* __restrict__ target,
                                                    const int* __restrict__ mask,
                                                    const int* __restrict__ groups,
                                                    unsigned int* __restrict__ ws) {
  __shared__ unsigned int lh[G_ * BINS1];   // 16 groups x 1024 bins = 64 KB
  unsigned int t = threadIdx.x;
#pragma unroll
  for (int i = 0; i < (G_ * BINS1) / 256; ++i) lh[t + i * 256] = 0u;
  __syncthreads();

  unsigned int b = blockIdx.x >> 5;         // 16 batches
  unsigned int chunk = blockIdx.x & 31u;    // 32 chunks of 4096 points
  unsigned int base = b * N_ + chunk * 4096u;
#pragma unroll
  for (int i = 0; i < 16; ++i) {
    unsigned int p = base + t + (unsigned int)i * 256u;
    if (mask[p]) {
      unsigned int g = (unsigned int)groups[p];
      unsigned int key = fkey(target[p * 3u + 2u]);
      atomicAdd(&lh[(g << 10) + (key >> 22)], 1u);
    }
  }
  __syncthreads();
  unsigned int* h1 = ws + OFF_H1 + (b << 14);   // b * 16 * 1024
#pragma unroll
  for (int i = 0; i < (G_ * BINS1) / 256; ++i) {
    unsigned int idx = t + (unsigned int)i * 256u;
    unsigned int c = lh[idx];
    if (c) atomicAdd(&h1[idx], c);
  }
}

// Select coarse bin containing rank k = (cnt-1)/2; record residual rank.
__global__ __launch_bounds__(256) void select1_kernel(unsigned int* __restrict__ ws) {
  __shared__ unsigned int pre[256];
  __shared__ unsigned int totS;
  unsigned int t = threadIdx.x, bg = blockIdx.x;
  const unsigned int* row = ws + OFF_H1 + bg * BINS1;
  unsigned int v4[4], s = 0u;
#pragma unroll
  for (int j = 0; j < 4; ++j) { v4[j] = row[t * 4u + j]; s += v4[j]; }
  pre[t] = s;
  __syncthreads();
  if (t == 0) {
    unsigned int acc = 0u;
    for (int j = 0; j < 256; ++j) { unsigned int v = pre[j]; pre[j] = acc; acc += v; }
    totS = acc;
    ws[OFF_CNT + bg] = acc;
    if (acc == 0u) { ws[OFF_SEL1 + bg] = 0xFFFFFFFFu; ws[OFF_REM1 + bg] = 0u; }
  }
  __syncthreads();
  unsigned int tot = totS;
  if (tot) {
    unsigned int k = (tot - 1u) >> 1;       // lower-median rank
    unsigned int p0 = pre[t];
    if (k >= p0 && k < p0 + s) {
      unsigned int r = k - p0, acc = 0u;
#pragma unroll
      for (int j = 0; j < 4; ++j) {
        if (r < acc + v4[j]) { ws[OFF_SEL1 + bg] = t * 4u + j; ws[OFF_REM1 + bg] = r - acc; break; }
        acc += v4[j];
      }
    }
  }
}

// Pass 2: histogram bits [21:11] for points whose top-10 bits match selected bin.
__global__ __launch_bounds__(256) void hist2_kernel(const float* __restrict__ target,
                                                    const int* __restrict__ mask,
                                                    const int* __restrict__ groups,
                                                    unsigned int* __restrict__ ws) {
  unsigned int p0 = blockIdx.x * 1024u + threadIdx.x;
#pragma unroll
  for (int i = 0; i < 4; ++i) {
    unsigned int p = p0 + (unsigned int)i * 256u;
    if (mask[p]) {
      unsigned int b = p >> 17;
      unsigned int bg = (b << 4) | (unsigned int)groups[p];
      unsigned int s1 = ws[OFF_SEL1 + bg];
      unsigned int key = fkey(target[p * 3u + 2u]);
      if ((key >> 22) == s1)
        atomicAdd(&ws[OFF_H2 + (bg << 11) + ((key >> 11) & 2047u)], 1u);
    }
  }
}

__global__ __launch_bounds__(256) void select2_kernel(unsigned int* __restrict__ ws) {
  __shared__ unsigned int pre[256];
  unsigned int t = threadIdx.x, bg = blockIdx.x;
  unsigned int s1 = ws[OFF_SEL1 + bg];
  if (s1 == 0xFFFFFFFFu) {                  // uniform per block
    if (t == 0) { ws[OFF_PFX + bg] = 0xFFFFFFFFu; ws[OFF_REM2 + bg] = 0u; }
    return;
  }
  const unsigned int* row = ws + OFF_H2 + (bg << 11);
  unsigned int v8[8], s = 0u;
#pragma unroll
  for (int j = 0; j < 8; ++j) { v8[j] = row[t * 8u + j]; s += v8[j]; }
  pre[t] = s;
  __syncthreads();
  if (t == 0) {
    unsigned int acc = 0u;
    for (int j = 0; j < 256; ++j) { unsigned int v = pre[j]; pre[j] = acc; acc += v; }
  }
  __syncthreads();
  unsigned int k = ws[OFF_REM1 + bg];
  unsigned int p0 = pre[t];
  if (k >= p0 && k < p0 + s) {
    unsigned int r = k - p0, acc = 0u;
#pragma unroll
    for (int j = 0; j < 8; ++j) {
      if (r < acc + v8[j]) { ws[OFF_PFX + bg] = (s1 << 11) | (t * 8u + j); ws[OFF_REM2 + bg] = r - acc; break; }
      acc += v8[j];
    }
  }
}

// Pass 3: histogram low 11 bits for points whose top-21 bits match prefix.
__global__ __launch_bounds__(256) void hist3_kernel(const float* __restrict__ target,
                                                    const int* __restrict__ mask,
                                                    const int* __restrict__ groups,
                                                    unsigned int* __restrict__ ws) {
  unsigned int p0 = blockIdx.x * 1024u + threadIdx.x;
#pragma unroll
  for (int i = 0; i < 4; ++i) {
    unsigned int p = p0 + (unsigned int)i * 256u;
    if (mask[p]) {
      unsigned int b = p >> 17;
      unsigned int bg = (b << 4) | (unsigned int)groups[p];
      unsigned int pfx = ws[OFF_PFX + bg];  // sentinel never matches (key>>11 < 2^21)
      unsigned int key = fkey(target[p * 3u + 2u]);
      if ((key >> 11) == pfx)
        atomicAdd(&ws[OFF_H3 + (bg << 11) + (key & 2047u)], 1u);
    }
  }
}

__global__ __launch_bounds__(256) void select3_kernel(unsigned int* __restrict__ ws) {
  __shared__ unsigned int pre[256];
  unsigned int t = threadIdx.x, bg = blockIdx.x;
  unsigned int pfx = ws[OFF_PFX + bg];
  float* invp = (float*)(ws + OFF_INV);
  if (pfx == 0xFFFFFFFFu) {                 // empty group -> median 1.0 -> inv 1.0
    if (t == 0) invp[bg] = 1.0f;
    return;
  }
  const unsigned int* row = ws + OFF_H3 + (bg << 11);
  unsigned int v8[8], s = 0u;
#pragma unroll
  for (int j = 0; j < 8; ++j) { v8[j] = row[t * 8u + j]; s += v8[j]; }
  pre[t] = s;
  __syncthreads();
  if (t == 0) {
    unsigned int acc = 0u;
    for (int j = 0; j < 256; ++j) { unsigned int v = pre[j]; pre[j] = acc; acc += v; }
  }
  __syncthreads();
  unsigned int k = ws[OFF_REM2 + bg];
  unsigned int p0 = pre[t];
  if (k >= p0 && k < p0 + s) {
    unsigned int r = k - p0, acc = 0u;
#pragma unroll
    for (int j = 0; j < 8; ++j) {
      if (r < acc + v8[j]) {
        unsigned int key = (pfx << 11) | (t * 8u + j);
        float med = fval(key);
        invp[bg] = 1.0f / fmaxf(fabsf(med), 1e-6f);
        break;
      }
      acc += v8[j];
    }
  }
}

// Final pass: masked log-L1 loss; wave reduction done on the matrix unit
// (v_wmma_f32_16x16x4_f32 with B = ones -> exact f32 row sums).
__global__ __launch_bounds__(256) void loss_kernel(const float* __restrict__ pred,
                                                   const float* __restrict__ target,
                                                   const int* __restrict__ mask,
                                                   const int* __restrict__ groups,
                                                   const unsigned int* __restrict__ ws,
                                                   float* __restrict__ bs) {
  unsigned int t = threadIdx.x;
  unsigned int p0 = blockIdx.x * 1024u + t;
  const float* invn = (const float*)(ws + OFF_INV);
  float ls = 0.0f;
#pragma unroll
  for (int i = 0; i < 4; ++i) {
    unsigned int p = p0 + (unsigned int)i * 256u;
    if (mask[p]) {
      unsigned int b = p >> 17;
      unsigned int bg = (b << 4) | (unsigned int)groups[p];
      float inv = invn[bg];
      unsigned int e = p * 3u;
      float px = pred[e] * inv, py = pred[e + 1u] * inv, pz = pred[e + 2u] * inv;
      float tx = target[e] * inv, ty = target[e + 1u] * inv, tz = target[e + 2u] * inv;
      ls += fabsf(logt(px) - logt(tx)) + fabsf(logt(py) - logt(ty)) + fabsf(logt(pz) - logt(tz));
    }
  }
  // Wave-level sum via WMMA: A(16x4)=per-lane partials, B(4x16)=ones, D=A*B.
  // EXEC is full here (no early returns in this kernel).
  v2f a; a.x = ls; a.y = 0.0f;
  v2f bm; bm.x = 1.0f; bm.y = 1.0f;
  v8f c = {};
  c = __builtin_amdgcn_wmma_f32_16x16x4_f32(false, a, false, bm, (short)0, c, false, false);
  float w = c[0] + c[1] + c[2] + c[3] + c[4] + c[5] + c[6] + c[7]; // rows 0-7 (lo lanes) / 8-15 (hi lanes)
  w += __shfl_xor(w, 16, 32);                                     // full wave sum in every lane

  __shared__ float wsum[8];
  unsigned int wave = t >> 5;
  if ((t & 31u) == 0u) wsum[wave] = w;
  __syncthreads();
  if (t == 0) {
    float s = 0.0f;
#pragma unroll
    for (int j = 0; j < 8; ++j) s += wsum[j];   // fixed order -> deterministic
    bs[blockIdx.x] = s;
  }
}

__global__ __launch_bounds__(256) void finalize_kernel(const unsigned int* __restrict__ ws,
                                                       const float* __restrict__ bs,
                                                       float* __restrict__ out) {
  __shared__ float sf[256];
  __shared__ unsigned int su[256];
  unsigned int t = threadIdx.x;
  float s = 0.0f;
#pragma unroll
  for (int i = 0; i < 8; ++i) s += bs[(unsigned int)i * 256u + t];  // fixed order
  sf[t] = s;
  su[t] = ws[OFF_CNT + t];          // total valid = sum of per-(b,g) counts
  __syncthreads();
  for (int off = 128; off > 0; off >>= 1) {
    if (t < (unsigned int)off) { sf[t] += sf[t + off]; su[t] += su[t + off]; }
    __syncthreads();
  }
  if (t == 0) out[0] = sf[0] / (3.0f * (float)su[0] + 1e-6f);
}

extern "C" void kernel_launch(void* const* d_in, const int* in_sizes, int n_in,
                              void* d_out, int out_size, void* d_ws, size_t ws_size,
                              hipStream_t stream) {
  const float* pred = (const float*)d_in[0];
  const float* target = (const float*)d_in[1];
  const int* mask = (const int*)d_in[2];
  const int* groups = (const int*)d_in[3];
  unsigned int* ws = (unsigned int*)d_ws;
  float* bs = (float*)(ws + OFF_BS);
  float* out = (float*)d_out;

  zero_ws_kernel<<<1280, 256, 0, stream>>>((uint4*)d_ws);                       // 5 MB hist clear
  hist1_kernel<<<512, 256, 0, stream>>>(target, mask, groups, ws);              // HBM pass (L2-fills)
  select1_kernel<<<256, 256, 0, stream>>>(ws);
  hist2_kernel<<<2048, 256, 0, stream>>>(target, mask, groups, ws);             // L2 pass
  select2_kernel<<<256, 256, 0, stream>>>(ws);
  hist3_kernel<<<2048, 256, 0, stream>>>(target, mask, groups, ws);             // L2 pass
  select3_kernel<<<256, 256, 0, stream>>>(ws);
  loss_kernel<<<2048, 256, 0, stream>>>(pred, target, mask, groups, ws, bs);    // pred from HBM, rest L2
  finalize_kernel<<<1, 256, 0, stream>>>(ws, bs, out);
}